// CrossAttention_56092272886201
// MI455X (gfx1250) — compile-verified
//
#include <hip/hip_runtime.h>

typedef float v2f __attribute__((ext_vector_type(2)));
typedef float v8f __attribute__((ext_vector_type(8)));
typedef int   v4i __attribute__((ext_vector_type(4)));

#define HEADS 8
#define DIM   64
#define NCTX  2048
#define TJ    16            // j-tile (WMMA N)
#define NT    (NCTX / TJ)   // 128 j-tiles
#define ITILES 2
#define IROWS (ITILES * 16) // 32 query rows per workgroup
#define LNEPS 1e-5f
// 1/sqrt(64) * log2(e): softmax in base-2 (v_exp_f32 is natively 2^x)
#define QSCALE 0.1803368801111244f

#if __has_builtin(__builtin_amdgcn_global_load_async_to_lds_b128)
#define HAVE_ASYNC_LDS 1
#else
#define HAVE_ASYNC_LDS 0
#endif

typedef __attribute__((address_space(1))) v4i* gv4p;   // global int4*
typedef __attribute__((address_space(3))) v4i* lv4p;   // LDS int4*

// Begin a 16-byte global->LDS tile move. Async path: DMA straight to LDS,
// tracked by ASYNCcnt. Fallback: load to regs now, store to LDS in _end()
// (software pipelining so s_wait_loadcnt lands after the compute).
static __device__ __forceinline__ void tile_ld_begin(const float* g, float* l,
                                                     float4* stage) {
#if HAVE_ASYNC_LDS
  __builtin_amdgcn_global_load_async_to_lds_b128(
      (gv4p)(uintptr_t)g, (lv4p)(unsigned)(uintptr_t)l, 0, 0);
  (void)stage;
#else
  *stage = *(const float4*)g;
  (void)l;
#endif
}
static __device__ __forceinline__ void tile_ld_end(float* l, const float4* stage) {
#if !HAVE_ASYNC_LDS
  l[0] = stage->x; l[1] = stage->y; l[2] = stage->z; l[3] = stage->w;
#else
  (void)l; (void)stage;
#endif
}
static __device__ __forceinline__ void tile_ld_wait() {
#if HAVE_ASYNC_LDS
#if __has_builtin(__builtin_amdgcn_s_wait_asynccnt)
  __builtin_amdgcn_s_wait_asynccnt(0);
#else
  asm volatile("s_wait_asynccnt 0x0" ::: "memory");
#endif
#endif
}

// ---- wave32 16-lane xor-butterfly via ds_swizzle (group-of-32 mode) ----
template<int XMASK>
static __device__ __forceinline__ float swzx(float v) {
  return __int_as_float(__builtin_amdgcn_ds_swizzle(__float_as_int(v), (XMASK << 10) | 0x1F));
}

// fp32 WMMA: D(16x16) = A(16x4) * B(4x16) + C
static __device__ __forceinline__ v8f wmma4(v2f a, v2f b, v8f c) {
  return __builtin_amdgcn_wmma_f32_16x16x4_f32(false, a, false, b, (short)0, c, false, false);
}

// ---------------------------------------------------------------------------
// Kernel 1: KV projection + per-head LayerNorm of V.  K,Vn -> workspace.
// ---------------------------------------------------------------------------
__global__ void ca_proj_kv(const float* __restrict__ ctx,
                           const float* __restrict__ Wkv,
                           const float* __restrict__ g_v,
                           const float* __restrict__ b_v,
                           float* __restrict__ Kws,
                           float* __restrict__ Vws) {
  __shared__ float ctx_l[DIM];
  __shared__ float wkv_l[2 * DIM * DIM];
  __shared__ float kvrow[2 * DIM];
  const int tid = threadIdx.x;      // 0..127 == output column m
  const int row = blockIdx.x;       // 0..4095 == (b*2048 + j)

  if (tid < DIM) ctx_l[tid] = ctx[(size_t)row * DIM + tid];
  for (int i = tid; i < 2 * DIM * DIM; i += 128) wkv_l[i] = Wkv[i];
  __syncthreads();

  float acc = 0.f;
#pragma unroll 8
  for (int d = 0; d < DIM; ++d) acc += ctx_l[d] * wkv_l[tid * DIM + d];
  kvrow[tid] = acc;
  __syncthreads();

  if (tid < DIM) {
    Kws[(size_t)row * DIM + tid] = kvrow[tid];
  } else {
    const int vi = tid - DIM;
    const int h  = vi >> 3;
    const float* vv = &kvrow[DIM + (h << 3)];
    float mu = 0.f;
#pragma unroll
    for (int e = 0; e < 8; ++e) mu += vv[e];
    mu *= 0.125f;
    float var = 0.f;
#pragma unroll
    for (int e = 0; e < 8; ++e) { float t = vv[e] - mu; var += t * t; }
    var *= 0.125f;
    const float rs = rsqrtf(var + LNEPS);
    const int e0 = vi & 7;
    Vws[(size_t)row * DIM + vi] = (kvrow[tid] - mu) * rs * g_v[e0] + b_v[e0];
  }
}

// ---------------------------------------------------------------------------
// Kernel 2: fused talking-heads attention, async double-buffered K/V tiles.
// One block = (b, 32-row i-tile); 8 waves = 8 heads.
// ---------------------------------------------------------------------------
__global__ void ca_attn(const float* __restrict__ x,
                        const float* __restrict__ Wq,
                        const float* __restrict__ W_talk,
                        const float* __restrict__ g_t,
                        const float* __restrict__ b_t,
                        const float* __restrict__ Kws,
                        const float* __restrict__ Vws,
                        float* __restrict__ out) {
  __shared__ float qs[IROWS][68];
  __shared__ float ks[2][TJ][68];                       // double-buffered
  __shared__ float vs[2][TJ][68];
  __shared__ float st[IROWS][17][8] __attribute__((aligned(32))); // probs [i][j][h]
  __shared__ float an[IROWS][17][8] __attribute__((aligned(32))); // mixed+LN [i][j][g]
  __shared__ float wt[HEADS * HEADS];
  __shared__ float gt[HEADS], bt[HEADS];

  const int tid  = threadIdx.x;
  const int lane = tid & 31;
  const int w    = tid >> 5;               // wave id == head id
  const int lr   = lane & 15;
  const int hi   = lane >> 4;

  const int bz = blockIdx.x >> 6;          // 64 i-tiles per batch
  const int i0 = (blockIdx.x & 63) * IROWS;
  const size_t rowbase = (size_t)bz * NCTX;

  // per-thread tile-load coordinates (one float4 per K/V tile)
  const int ldr = tid >> 4, ldc = (tid & 15) * 4;

  if (tid < 64)       wt[tid]      = W_talk[tid];
  else if (tid < 72)  gt[tid - 64] = g_t[tid - 64];
  else if (tid < 80)  bt[tid - 72] = b_t[tid - 72];

  // ---- stage x tile (reuse st as scratch) and project Q ----
  float* xstage = &st[0][0][0];            // needs 2048 floats, st has 4352
#pragma unroll
  for (int k = 0; k < 2; ++k) {
    const int idx = tid + 256 * k;
    const int r = idx >> 4, c4 = (idx & 15) * 4;
    const float4 xv = *(const float4*)&x[(rowbase + i0 + r) * DIM + c4];
    xstage[r * DIM + c4 + 0] = xv.x; xstage[r * DIM + c4 + 1] = xv.y;
    xstage[r * DIM + c4 + 2] = xv.z; xstage[r * DIM + c4 + 3] = xv.w;
  }
  __syncthreads();
#pragma unroll
  for (int rep = 0; rep < 8; ++rep) {
    const int idx = tid + 256 * rep;
    const int i = idx >> 6, m = idx & 63;
    float acc = 0.f;
#pragma unroll 8
    for (int d = 0; d < DIM; ++d) acc += xstage[i * DIM + d] * Wq[m * DIM + d];
    qs[i][m] = acc * QSCALE;
  }
  __syncthreads();

  // cache loop-invariant Q A-operands in registers
  v2f qa[ITILES][2];
#pragma unroll
  for (int it = 0; it < ITILES; ++it)
#pragma unroll
    for (int t = 0; t < 2; ++t) {
      const int kc = w * 8 + 4 * t + 2 * hi;
      qa[it][t].x = qs[it * 16 + lr][kc];
      qa[it][t].y = qs[it * 16 + lr][kc + 1];
    }

  // ================= pass 1: z = sum_j 2^s  (per-lane partials) =============
  float zL[ITILES][8];
#pragma unroll
  for (int it = 0; it < ITILES; ++it)
#pragma unroll
    for (int r = 0; r < 8; ++r) zL[it][r] = 0.f;

  {                                        // preload tile 0 -> buffer 0
    float4 stg;
    tile_ld_begin(&Kws[(rowbase + ldr) * DIM + ldc], &ks[0][ldr][ldc], &stg);
    tile_ld_end(&ks[0][ldr][ldc], &stg);
    tile_ld_wait();
  }
  __syncthreads();

  for (int jt = 0; jt < NT; ++jt) {
    const int cur = jt & 1;
    float4 kstg;
    const bool pre = (jt + 1 < NT);
    if (pre) {                             // kick off next tile early
      const size_t rb = (rowbase + (jt + 1) * TJ + ldr) * DIM + ldc;
      tile_ld_begin(&Kws[rb], &ks[cur ^ 1][ldr][ldc], &kstg);
    }
#pragma unroll
    for (int it = 0; it < ITILES; ++it) {
      v8f s = {0, 0, 0, 0, 0, 0, 0, 0};
#pragma unroll
      for (int t = 0; t < 2; ++t) {
        const int kc = w * 8 + 4 * t + 2 * hi;
        v2f b; b.x = ks[cur][lr][kc]; b.y = ks[cur][lr][kc + 1];
        s = wmma4(qa[it][t], b, s);
      }
#pragma unroll
      for (int r = 0; r < 8; ++r)
        zL[it][r] += __builtin_amdgcn_exp2f(s[r]);
    }
    if (pre) tile_ld_end(&ks[cur ^ 1][ldr][ldc], &kstg);
    tile_ld_wait();
    __syncthreads();                       // next buffer visible to all waves
  }
  // one-time 16-lane butterfly add of per-lane z partials
#pragma unroll
  for (int it = 0; it < ITILES; ++it)
#pragma unroll
    for (int r = 0; r < 8; ++r) {
      float z = zL[it][r];
      z += swzx<1>(z); z += swzx<2>(z); z += swzx<4>(z); z += swzx<8>(z);
      zL[it][r] = 1.f / z;                 // keep reciprocal for pass 2
    }

  // ========== pass 2: probs -> head-mix -> LN(heads) -> A*V ==========
  v8f oc[ITILES];
#pragma unroll
  for (int it = 0; it < ITILES; ++it) oc[it] = (v8f){0, 0, 0, 0, 0, 0, 0, 0};

  {                                        // preload tile 0 -> buffer 0
    const size_t rb = (rowbase + ldr) * DIM + ldc;
    float4 kstg, vstg;
    tile_ld_begin(&Kws[rb], &ks[0][ldr][ldc], &kstg);
    tile_ld_begin(&Vws[rb], &vs[0][ldr][ldc], &vstg);
    tile_ld_end(&ks[0][ldr][ldc], &kstg);
    tile_ld_end(&vs[0][ldr][ldc], &vstg);
    tile_ld_wait();
  }
  __syncthreads();

  for (int jt = 0; jt < NT; ++jt) {
    const int cur = jt & 1;
    float4 kstg, vstg;
    const bool pre = (jt + 1 < NT);
    if (pre) {
      const size_t rb = (rowbase + (jt + 1) * TJ + ldr) * DIM + ldc;
      tile_ld_begin(&Kws[rb], &ks[cur ^ 1][ldr][ldc], &kstg);
      tile_ld_begin(&Vws[rb], &vs[cur ^ 1][ldr][ldc], &vstg);
    }
#pragma unroll
    for (int it = 0; it < ITILES; ++it) {
      v8f s = {0, 0, 0, 0, 0, 0, 0, 0};
#pragma unroll
      for (int t = 0; t < 2; ++t) {
        const int kc = w * 8 + 4 * t + 2 * hi;
        v2f b; b.x = ks[cur][lr][kc]; b.y = ks[cur][lr][kc + 1];
        s = wmma4(qa[it][t], b, s);
      }
#pragma unroll
      for (int r = 0; r < 8; ++r)          // finalized probabilities
        st[it * 16 + r + 8 * hi][lr][w] =
            __builtin_amdgcn_exp2f(s[r]) * zL[it][r];
    }
    __syncthreads();                       // probs visible
#pragma unroll
    for (int k = 0; k < 2; ++k) {          // 512 (i,j) pairs, 2 per thread
      const int p = tid + 256 * k;
      const int i = p >> 4, j = p & 15;
      const v8f at = *(const v8f*)&st[i][j][0];   // ds_load_b128 x2
      float ag[8]; float mu = 0.f;
#pragma unroll
      for (int g = 0; g < 8; ++g) {
        float a = 0.f;
#pragma unroll
        for (int h = 0; h < 8; ++h) a += at[h] * wt[g * 8 + h];
        ag[g] = a; mu += a;
      }
      mu *= 0.125f;
      float var = 0.f;
#pragma unroll
      for (int g = 0; g < 8; ++g) { float t = ag[g] - mu; var += t * t; }
      var *= 0.125f;
      const float rs = rsqrtf(var + LNEPS);
      v8f res;
#pragma unroll
      for (int g = 0; g < 8; ++g) res[g] = (ag[g] - mu) * rs * gt[g] + bt[g];
      *(v8f*)&an[i][j][0] = res;                  // ds_store_b128 x2
    }
    __syncthreads();                       // mixed weights visible
#pragma unroll
    for (int it = 0; it < ITILES; ++it)
#pragma unroll
      for (int t = 0; t < 4; ++t) {        // contract over j-tile (K=16)
        const int kj = 4 * t + 2 * hi;
        v2f a; a.x = an[it * 16 + lr][kj][w]; a.y = an[it * 16 + lr][kj + 1][w];
        v2f b;
        b.x = (lr < 8) ? vs[cur][kj][w * 8 + lr]     : 0.f;
        b.y = (lr < 8) ? vs[cur][kj + 1][w * 8 + lr] : 0.f;
        oc[it] = wmma4(a, b, oc[it]);
      }
    if (pre) {
      tile_ld_end(&ks[cur ^ 1][ldr][ldc], &kstg);
      tile_ld_end(&vs[cur ^ 1][ldr][ldc], &vstg);
    }
    tile_ld_wait();
    __syncthreads();                       // next buffer + st/an reuse safe
  }

  // ---- write out[b, i, h*dv + d] ----
  if (lr < 8) {
    const int col = w * 8 + lr;
#pragma unroll
    for (int it = 0; it < ITILES; ++it)
#pragma unroll
      for (int r = 0; r < 8; ++r) {
        const int row = i0 + it * 16 + r + 8 * hi;
        out[(rowbase + row) * DIM + col] = oc[it][r];
      }
  }
}

extern "C" void kernel_launch(void* const* d_in, const int* in_sizes, int n_in,
                              void* d_out, int out_size, void* d_ws, size_t ws_size,
                              hipStream_t stream) {
  const float* x    = (const float*)d_in[0];
  const float* ctx  = (const float*)d_in[1];
  const float* Wq   = (const float*)d_in[2];
  const float* Wkv  = (const float*)d_in[3];
  const float* g_v  = (const float*)d_in[4];
  const float* b_v  = (const float*)d_in[5];
  const float* W_tk = (const float*)d_in[6];
  const float* g_t  = (const float*)d_in[7];
  const float* b_t  = (const float*)d_in[8];
  float* out = (float*)d_out;

  float* Kws = (float*)d_ws;                 // 1 MB
  float* Vws = Kws + (size_t)2 * NCTX * DIM; // 1 MB

  ca_proj_kv<<<2 * NCTX, 128, 0, stream>>>(ctx, Wkv, g_v, b_v, Kws, Vws);
  ca_attn<<<2 * (NCTX / IROWS), 256, 0, stream>>>(x, Wq, W_tk, g_t, b_t, Kws, Vws, out);
}